// TanhNewtonLayer_412316860504
// MI455X (gfx1250) — compile-verified
//
#include <hip/hip_runtime.h>

// ---------------------------------------------------------------------------
// z = tanh(W z + x) fixed-point solver for MI455X (gfx1250, wave32).
//
// Per 16-row tile (256 blocks -> fills the chip), each of the 8 waves owns a
// 16x16 column tile of the [16,128] state. B fragments of W^T are loaded ONCE
// from global W (contiguous float2 per lane, L2-resident) into 64 VGPRs.
// The iteration loop is {ds_load_b64 A-frag -> v_wmma_f32_16x16x4_f32} over
// FOUR independent accumulator chains (kk mod 4) so the matrix pipe issues at
// throughput instead of stalling on the D->C RAW latency of a single chain.
// All state (Z, X) lives in ~16.7 KB LDS; HBM sees one read of x/W and one
// masked write of z. 41 GEMM passes fully on-chip.
// ---------------------------------------------------------------------------

typedef float v2f __attribute__((ext_vector_type(2)));
typedef float v8f __attribute__((ext_vector_type(8)));

#define DDIM   128
#define M_TILE 16
#define NITER  40                  // Picard passes; rho(sech2*W)^40 << TOL
#define TOLSQ  (1e-4f * 1e-4f)
#define ZS     132                 // Z row stride: (4m+k) mod 64 -> 0 conflicts

__global__ __launch_bounds__(256, 1)
void tanh_fixed_point_kernel(const float* __restrict__ x,
                             const float* __restrict__ W,
                             float* __restrict__ out) {
    __shared__ float Z[M_TILE * ZS];      // 8448 B
    __shared__ float X[M_TILE * DDIM];    // 8192 B
    __shared__ float RN[M_TILE];          // residual row-norms^2

    const int tid  = threadIdx.x;         // 256 threads = 8 waves (wave32)
    const int row0 = blockIdx.x * M_TILE;

    // ---- one-time: X tile -> LDS, init Z = tanh(X) ----
    for (int idx = tid; idx < M_TILE * DDIM; idx += 256) {
        const int m = idx >> 7, k = idx & (DDIM - 1);
        const float xv = x[(row0 + m) * DDIM + k];
        X[m * DDIM + k] = xv;
        Z[m * ZS + k]   = tanhf(xv);
    }
    if (tid < M_TILE) RN[tid] = 0.0f;

    const int wid  = tid >> 5;            // wave -> column tile
    const int lane = tid & 31;
    const int half = lane >> 4;           // selects K pair (B) / M+8 (C)
    const int l16  = lane & 15;
    const int n0   = wid << 4;

    // ---- B fragments of W^T, straight from global W into registers ----
    // Lane needs (W^T[kb][n], W^T[kb+1][n]) = (W[n][kb], W[n][kb+1]): a
    // contiguous float2 in row-major W. One-time load, L2-resident.
    v2f bfrag[DDIM / 4];
    {
        const float* wr = W + (n0 + l16) * DDIM + (half << 1);
#pragma unroll
        for (int kk = 0; kk < DDIM / 4; ++kk)
            bfrag[kk] = *reinterpret_cast<const v2f*>(wr + (kk << 2));
    }
    __syncthreads();

    const float* za = &Z[l16 * ZS + (half << 1)];  // per-lane A-frag base

    // ---- fixed-point iterations (+1 residual pass), all on-chip ----
    for (int it = 0; it <= NITER; ++it) {
        // 4 independent accumulator chains -> matrix pipe runs at issue rate
        v8f acc0 = {}, acc1 = {}, acc2 = {}, acc3 = {};
#pragma unroll
        for (int kk = 0; kk < DDIM / 4; kk += 4) {
            const v2f a0 = *reinterpret_cast<const v2f*>(za + ((kk + 0) << 2));
            const v2f a1 = *reinterpret_cast<const v2f*>(za + ((kk + 1) << 2));
            const v2f a2 = *reinterpret_cast<const v2f*>(za + ((kk + 2) << 2));
            const v2f a3 = *reinterpret_cast<const v2f*>(za + ((kk + 3) << 2));
            acc0 = __builtin_amdgcn_wmma_f32_16x16x4_f32(
                false, a0, false, bfrag[kk + 0], (short)0, acc0, false, false);
            acc1 = __builtin_amdgcn_wmma_f32_16x16x4_f32(
                false, a1, false, bfrag[kk + 1], (short)0, acc1, false, false);
            acc2 = __builtin_amdgcn_wmma_f32_16x16x4_f32(
                false, a2, false, bfrag[kk + 2], (short)0, acc2, false, false);
            acc3 = __builtin_amdgcn_wmma_f32_16x16x4_f32(
                false, a3, false, bfrag[kk + 3], (short)0, acc3, false, false);
        }
        __syncthreads();                   // all Z reads done before rewrite

        if (it < NITER) {
            // Z = tanh(Z@W^T + X); C layout: VGPR e -> row e (+8 for half=1)
#pragma unroll
            for (int e = 0; e < 8; ++e) {
                const int m  = e + (half << 3);
                const int n  = n0 + l16;
                const float zl = ((acc0[e] + acc1[e]) + (acc2[e] + acc3[e]))
                                 + X[m * DDIM + n];
                Z[m * ZS + n] = tanhf(zl);
            }
        } else {
            // residual: g = z - tanh(z@W^T + x); row norm via wave32 shuffles
#pragma unroll
            for (int e = 0; e < 8; ++e) {
                const int m  = e + (half << 3);
                const int n  = n0 + l16;
                const float zl = ((acc0[e] + acc1[e]) + (acc2[e] + acc3[e]))
                                 + X[m * DDIM + n];
                const float g = Z[m * ZS + n] - tanhf(zl);
                float s = g * g;
                s += __shfl_xor(s, 1);
                s += __shfl_xor(s, 2);
                s += __shfl_xor(s, 4);
                s += __shfl_xor(s, 8);     // sum over 16 columns of row m
                if (l16 == 0) atomicAdd(&RN[m], s);   // 1 atomic / row / wave
            }
        }
        __syncthreads();
    }

    // ---- masked, vectorized store: zero rows with ||g|| > TOL ----
    for (int idx = tid; idx < M_TILE * (DDIM / 4); idx += 256) {
        const int m  = idx >> 5;           // 32 float4 per row
        const int c4 = idx & 31;
        float4 zv = *reinterpret_cast<const float4*>(&Z[m * ZS + (c4 << 2)]);
        if (RN[m] > TOLSQ) zv = make_float4(0.f, 0.f, 0.f, 0.f);
        *reinterpret_cast<float4*>(&out[(row0 + m) * DDIM + (c4 << 2)]) = zv;
    }
}

extern "C" void kernel_launch(void* const* d_in, const int* in_sizes, int n_in,
                              void* d_out, int out_size, void* d_ws, size_t ws_size,
                              hipStream_t stream) {
    (void)n_in; (void)out_size; (void)d_ws; (void)ws_size;
    const float* x = (const float*)d_in[0];   // [B, 128] f32
    const float* W = (const float*)d_in[1];   // [128, 128] f32
    float* out = (float*)d_out;               // [B, 128] f32

    const int rows   = in_sizes[0] / DDIM;    // B = 4096
    const int blocks = rows / M_TILE;         // 256 workgroups
    tanh_fixed_point_kernel<<<blocks, 256, 0, stream>>>(x, W, out);
}